// GNNClassifier_68289980006470
// MI455X (gfx1250) — compile-verified
//
#include <hip/hip_runtime.h>
#include <hip/hip_bf16.h>
#include <math.h>

typedef __attribute__((ext_vector_type(2))) float v2f;
typedef __attribute__((ext_vector_type(8))) float v8f;

#define HID 64
#define NGRAPHS 256

// ---------------- degree kernels ----------------
__global__ void deg_init_kernel(float* __restrict__ deg, int n) {
  int i = blockIdx.x * blockDim.x + threadIdx.x;
  if (i < n) deg[i] = 1.0f;  // self-loop contributes 1 to every node's degree
}

__global__ void deg_edge_kernel(const int* __restrict__ ei, float* __restrict__ deg, int E) {
  int e = blockIdx.x * blockDim.x + threadIdx.x;
  if (e < E) unsafeAtomicAdd(&deg[ei[E + e]], 1.0f);  // dst row of edge_index
}

__global__ void deg_to_dis_kernel(float* __restrict__ deg, int n) {
  int i = blockIdx.x * blockDim.x + threadIdx.x;
  if (i < n) deg[i] = rsqrtf(deg[i]);  // deg >= 1 always (self-loops)
}

// ---------------- WMMA GEMM: Y[n_rows,64] = X[n_rows,K] @ W[K,64] ----------------
// One wave handles a 16-row slab and all 4 N-tiles of 16.
// A 16x4 f32 layout: lanes 0-15 hold M=0..15 with K=(0,1); lanes 16-31 K=(2,3).
// C/D: VGPR v holds M=v (lanes 0-15) / M=v+8 (lanes 16-31), N=lane%16.
template <int K>
__global__ __launch_bounds__(256) void gemm_xw_kernel(const float* __restrict__ X,
                                                      const float* __restrict__ W,
                                                      float* __restrict__ Y, int n_rows) {
  int wave = (int)((blockIdx.x * blockDim.x + threadIdx.x) >> 5);
  int rowBase = wave * 16;
  if (rowBase >= n_rows) return;  // whole-wave exit keeps EXEC all-ones for WMMA
  int lane = threadIdx.x & 31;
  int hlf = lane >> 4;   // 0 or 1
  int l16 = lane & 15;

  const float* xrow = X + (size_t)(rowBase + l16) * K;
  v8f acc0 = {}, acc1 = {}, acc2 = {}, acc3 = {};

  for (int k = 0; k < K; k += 4) {
    int ka = k + hlf * 2;
    v2f a;
    a.x = xrow[ka];
    a.y = xrow[ka + 1];
    const float* w0 = W + (size_t)ka * HID + l16;        // row K=ka
    const float* w1 = W + (size_t)(ka + 1) * HID + l16;  // row K=ka+1
    v2f b0 = {w0[0],  w1[0]};
    v2f b1 = {w0[16], w1[16]};
    v2f b2 = {w0[32], w1[32]};
    v2f b3 = {w0[48], w1[48]};
    acc0 = __builtin_amdgcn_wmma_f32_16x16x4_f32(false, a, false, b0, (short)0, acc0, false, false);
    acc1 = __builtin_amdgcn_wmma_f32_16x16x4_f32(false, a, false, b1, (short)0, acc1, false, false);
    acc2 = __builtin_amdgcn_wmma_f32_16x16x4_f32(false, a, false, b2, (short)0, acc2, false, false);
    acc3 = __builtin_amdgcn_wmma_f32_16x16x4_f32(false, a, false, b3, (short)0, acc3, false, false);
  }

  float* yrow = Y + (size_t)(rowBase + hlf * 8) * HID + l16;
#pragma unroll
  for (int vg = 0; vg < 8; ++vg) {
    yrow[(size_t)vg * HID + 0]  = acc0[vg];
    yrow[(size_t)vg * HID + 16] = acc1[vg];
    yrow[(size_t)vg * HID + 32] = acc2[vg];
    yrow[(size_t)vg * HID + 48] = acc3[vg];
  }
}

// ---------------- edge scatter: out[dst] += dis[src]*dis[dst]*xw[src] ----------------
// One wave per edge; each lane handles 2 contiguous floats (coalesced 256B gather).
__global__ __launch_bounds__(256) void aggregate_kernel(const float* __restrict__ xw,
                                                        const int* __restrict__ ei,
                                                        const float* __restrict__ dis,
                                                        float* __restrict__ out, int E) {
  int e = (int)((blockIdx.x * blockDim.x + threadIdx.x) >> 5);
  if (e >= E) return;
  int lane = threadIdx.x & 31;
  int s = ei[e];
  int d = ei[E + e];
  float norm = dis[s] * dis[d];
  const v2f* src = (const v2f*)(xw + (size_t)s * HID);
  v2f v = src[lane];
  float* o = out + (size_t)d * HID + lane * 2;
  unsafeAtomicAdd(o, norm * v.x);
  unsafeAtomicAdd(o + 1, norm * v.y);
}

// ---------------- bias + self-loop + optional ReLU (in place on agg) ----------------
__global__ void bias_self_kernel(float* __restrict__ agg, const float* __restrict__ xw,
                                 const float* __restrict__ dis, const float* __restrict__ bias,
                                 int n, int do_relu) {
  int idx = blockIdx.x * blockDim.x + threadIdx.x;
  if (idx >= n * HID) return;
  int i = idx >> 6;
  int c = idx & 63;
  float di = dis[i];
  float v = agg[idx] + di * di * xw[idx] + bias[c];
  if (do_relu) v = fmaxf(v, 0.0f);
  agg[idx] = v;
}

// ---------------- graph mean-pool accumulation ----------------
__global__ __launch_bounds__(256) void pool_kernel(const float* __restrict__ h,
                                                   const int* __restrict__ batch,
                                                   float* __restrict__ sums,
                                                   float* __restrict__ counts, int n) {
  int i = (int)((blockIdx.x * blockDim.x + threadIdx.x) >> 5);
  if (i >= n) return;
  int lane = threadIdx.x & 31;
  int g = batch[i];
  const v2f* src = (const v2f*)(h + (size_t)i * HID);
  v2f v = src[lane];
  float* o = sums + (size_t)g * HID + lane * 2;
  unsafeAtomicAdd(o, v.x);
  unsafeAtomicAdd(o + 1, v.y);
  if (lane == 0) unsafeAtomicAdd(&counts[g], 1.0f);
}

// ---------------- MLP head: relu(pooled@Wf1+bf1)@Wf2+bf2, one wave per graph --------
__global__ __launch_bounds__(32) void head_kernel(const float* __restrict__ sums,
                                                  const float* __restrict__ counts,
                                                  const float* __restrict__ Wf1,
                                                  const float* __restrict__ bf1,
                                                  const float* __restrict__ Wf2,
                                                  const float* __restrict__ bf2,
                                                  float* __restrict__ out) {
  __shared__ float pooled[HID];
  __shared__ float t[32];
  int g = blockIdx.x;
  int lane = threadIdx.x;
  float inv = 1.0f / fmaxf(counts[g], 1.0f);
  pooled[lane]      = sums[(size_t)g * HID + lane] * inv;
  pooled[lane + 32] = sums[(size_t)g * HID + lane + 32] * inv;
  __syncthreads();
  float acc = bf1[lane];
#pragma unroll 8
  for (int k = 0; k < HID; ++k) acc += pooled[k] * Wf1[k * 32 + lane];
  t[lane] = fmaxf(acc, 0.0f);
  __syncthreads();
  if (lane < 8) {
    float o = bf2[lane];
#pragma unroll
    for (int j = 0; j < 32; ++j) o += t[j] * Wf2[j * 8 + lane];
    out[g * 8 + lane] = o;
  }
}

extern "C" void kernel_launch(void* const* d_in, const int* in_sizes, int n_in,
                              void* d_out, int out_size, void* d_ws, size_t ws_size,
                              hipStream_t stream) {
  const float* x     = (const float*)d_in[0];
  const int*   ei    = (const int*)d_in[1];   // [2, E] flattened: row0=src, row1=dst
  const int*   batch = (const int*)d_in[2];
  const float* W1  = (const float*)d_in[3];
  const float* b1  = (const float*)d_in[4];
  const float* W2  = (const float*)d_in[5];
  const float* b2  = (const float*)d_in[6];
  const float* Wf1 = (const float*)d_in[7];
  const float* bf1 = (const float*)d_in[8];
  const float* Wf2 = (const float*)d_in[9];
  const float* bf2 = (const float*)d_in[10];
  float* out = (float*)d_out;

  int n = in_sizes[0] / 128;  // N_NODES
  int E = in_sizes[1] / 2;    // N_EDGES

  // workspace layout (floats)
  float* dis = (float*)d_ws;
  size_t nAlign = ((size_t)n + 255) & ~(size_t)255;
  float* bufA   = dis + nAlign;               // xw (gemm output)
  float* bufB   = bufA + (size_t)n * HID;     // aggregation target / h
  float* sums   = bufB + (size_t)n * HID;     // [G, 64]
  float* counts = sums + (size_t)NGRAPHS * HID;

  int tb = 256;
  int nBlk = (n + tb - 1) / tb;
  int eBlk = (E + tb - 1) / tb;
  int waveBlkN = (n + 8 * 16 - 1) / (8 * 16);  // 8 waves/block, 16 rows/wave
  int edgeWaveBlk = (E + 7) / 8;               // 8 edges/block (1 wave each)
  int nodeWaveBlk = (n + 7) / 8;
  int nhBlk = ((n * HID) + tb - 1) / tb;

  // symmetric normalization dis = 1/sqrt(deg)
  deg_init_kernel<<<nBlk, tb, 0, stream>>>(dis, n);
  deg_edge_kernel<<<eBlk, tb, 0, stream>>>(ei, dis, E);
  deg_to_dis_kernel<<<nBlk, tb, 0, stream>>>(dis, n);

  // ---- layer 1 ----
  gemm_xw_kernel<128><<<waveBlkN, tb, 0, stream>>>(x, W1, bufA, n);
  hipMemsetAsync(bufB, 0, (size_t)n * HID * sizeof(float), stream);
  aggregate_kernel<<<edgeWaveBlk, tb, 0, stream>>>(bufA, ei, dis, bufB, E);
  bias_self_kernel<<<nhBlk, tb, 0, stream>>>(bufB, bufA, dis, b1, n, 1);

  // ---- layer 2 ----
  gemm_xw_kernel<64><<<waveBlkN, tb, 0, stream>>>(bufB, W2, bufA, n);
  hipMemsetAsync(bufB, 0, (size_t)n * HID * sizeof(float), stream);
  aggregate_kernel<<<edgeWaveBlk, tb, 0, stream>>>(bufA, ei, dis, bufB, E);
  bias_self_kernel<<<nhBlk, tb, 0, stream>>>(bufB, bufA, dis, b2, n, 0);

  // ---- pool + head ----
  hipMemsetAsync(sums, 0, (size_t)(NGRAPHS * HID + NGRAPHS) * sizeof(float), stream);
  pool_kernel<<<nodeWaveBlk, tb, 0, stream>>>(bufB, batch, sums, counts, n);
  head_kernel<<<NGRAPHS, 32, 0, stream>>>(sums, counts, Wf1, bf1, Wf2, bf2, out);
}